// GATConv_75711683494311
// MI455X (gfx1250) — compile-verified
//
#include <hip/hip_runtime.h>
#include <cstdint>
#include <cstddef>

static constexpr int   kN   = 100000;
static constexpr int   kE   = 1600000;
static constexpr float kNeg = 0.2f;
static constexpr float kEps = 1e-5f;

typedef __attribute__((ext_vector_type(2))) float v2f;
typedef __attribute__((ext_vector_type(8))) float v8f;

__device__ inline void atomic_max_f32(float* addr, float v) {
  // Sign-split trick: signed-int order matches float order for v>=0 and for
  // pos-vs-neg; reversed unsigned order matches for negatives. Init = -inf.
  if (v >= 0.0f) atomicMax(reinterpret_cast<int*>(addr), __float_as_int(v));
  else           atomicMin(reinterpret_cast<unsigned int*>(addr), __float_as_uint(v));
}

// ---- K0: wl[h*64+k] = sum_d W[k,h*64+d]*attn_l[h,d]  (and wr) -------------
__global__ void k_proj_vec(const float* __restrict__ W,
                           const float* __restrict__ attnl,
                           const float* __restrict__ attnr,
                           float* __restrict__ wl, float* __restrict__ wr) {
  const int t = threadIdx.x;            // 0..127
  const int h = t >> 6, k = t & 63;
  float sl = 0.f, sr = 0.f;
  #pragma unroll 8
  for (int d = 0; d < 64; ++d) {
    const float w = W[k * 128 + h * 64 + d];
    sl += w * attnl[h * 64 + d];
    sr += w * attnr[h * 64 + d];
  }
  wl[h * 64 + k] = sl;
  wr[h * 64 + k] = sr;
}

// ---- K_init: zero agg / bn stats, smax=-inf, denom=0 ----------------------
__global__ void k_init(float* __restrict__ smax, float* __restrict__ denom,
                       float* __restrict__ agg, float* __restrict__ bnsum) {
  const size_t i = (size_t)blockIdx.x * 256 + threadIdx.x;   // grid = N*128
  if (i < (size_t)kN * 128) agg[i] = 0.f;
  if (i < (size_t)kN * 2) {
    reinterpret_cast<int*>(smax)[i] = 0xFF800000;            // -inf
    denom[i] = 0.f;
  }
  if (i < 128) bnsum[i] = 0.f;                                // bnsum(64)+bnsq(64)
}

// ---- K1: per-node projections al[n,h]=atom[n]·wl_h, er[n,h]=atom[n]·wr_h --
__global__ void k_node(const float* __restrict__ atom,
                       const float* __restrict__ wl, const float* __restrict__ wr,
                       float* __restrict__ al, float* __restrict__ er) {
  __shared__ float sl[128], sr[128];
  const int t = threadIdx.x;
  if (t < 128) { sl[t] = wl[t]; sr[t] = wr[t]; }
  __syncthreads();
  const int n = blockIdx.x * blockDim.x + t;
  if (n >= kN) return;
  const float4* a4 = reinterpret_cast<const float4*>(atom + (size_t)n * 64);
  float a0 = 0, a1 = 0, e0 = 0, e1 = 0;
  #pragma unroll 4
  for (int q = 0; q < 16; ++q) {
    const float4 v = a4[q];
    const int k = q * 4;
    a0 += v.x * sl[k]      + v.y * sl[k + 1]      + v.z * sl[k + 2]      + v.w * sl[k + 3];
    a1 += v.x * sl[64 + k] + v.y * sl[64 + k + 1] + v.z * sl[64 + k + 2] + v.w * sl[64 + k + 3];
    e0 += v.x * sr[k]      + v.y * sr[k + 1]      + v.z * sr[k + 2]      + v.w * sr[k + 3];
    e1 += v.x * sr[64 + k] + v.y * sr[64 + k + 1] + v.z * sr[64 + k + 2] + v.w * sr[64 + k + 3];
  }
  al[(size_t)n * 2] = a0; al[(size_t)n * 2 + 1] = a1;
  er[(size_t)n * 2] = e0; er[(size_t)n * 2 + 1] = e1;
}

// ---- K2: edge logits s = leaky(al[src] + bond·wl + er[dst]); seg-max ------
__global__ void k_edge_score(const float* __restrict__ bond,
                             const int* __restrict__ src, const int* __restrict__ dst,
                             const float* __restrict__ wl,
                             const float* __restrict__ al, const float* __restrict__ er,
                             float* __restrict__ sx, float* __restrict__ smax) {
  __shared__ float sl[128];
  const int t = threadIdx.x;
  if (t < 128) sl[t] = wl[t];
  __syncthreads();
  const size_t e = (size_t)blockIdx.x * blockDim.x + t;
  if (e >= kE) return;
  __builtin_prefetch(bond + ((e + 4096 < kE) ? (e + 4096) : e) * 64, 0, 0);
  const float4* b4 = reinterpret_cast<const float4*>(bond + e * 64);
  float b0 = 0, b1 = 0;
  #pragma unroll 4
  for (int q = 0; q < 16; ++q) {
    const float4 v = b4[q];
    const int k = q * 4;
    b0 += v.x * sl[k]      + v.y * sl[k + 1]      + v.z * sl[k + 2]      + v.w * sl[k + 3];
    b1 += v.x * sl[64 + k] + v.y * sl[64 + k + 1] + v.z * sl[64 + k + 2] + v.w * sl[64 + k + 3];
  }
  const int s = src[e], d = dst[e];
  float s0 = al[(size_t)s * 2]     + b0 + er[(size_t)d * 2];
  float s1 = al[(size_t)s * 2 + 1] + b1 + er[(size_t)d * 2 + 1];
  s0 = (s0 > 0.f) ? s0 : kNeg * s0;
  s1 = (s1 > 0.f) ? s1 : kNeg * s1;
  sx[e * 2] = s0; sx[e * 2 + 1] = s1;
  atomic_max_f32(smax + (size_t)d * 2, s0);
  atomic_max_f32(smax + (size_t)d * 2 + 1, s1);
}

// ---- K3: ex = exp(s - smax[dst]); seg-sum denom ---------------------------
__global__ void k_edge_exp(const int* __restrict__ dst, float* __restrict__ sx,
                           const float* __restrict__ smax, float* __restrict__ denom) {
  const size_t e = (size_t)blockIdx.x * blockDim.x + threadIdx.x;
  if (e >= kE) return;
  const int d = dst[e];
  const float ex0 = __expf(sx[e * 2]     - smax[(size_t)d * 2]);
  const float ex1 = __expf(sx[e * 2 + 1] - smax[(size_t)d * 2 + 1]);
  sx[e * 2] = ex0; sx[e * 2 + 1] = ex1;
  atomicAdd(denom + (size_t)d * 2, ex0);
  atomicAdd(denom + (size_t)d * 2 + 1, ex1);
}

// ---- K4: agg[dst,h,:] += alpha_h * (atom[src] + bond[e]) ------------------
__global__ void k_edge_agg(const float* __restrict__ atom, const float* __restrict__ bond,
                           const int* __restrict__ src, const int* __restrict__ dst,
                           const float* __restrict__ sx, const float* __restrict__ denom,
                           float* __restrict__ agg) {
  const size_t e = (size_t)blockIdx.x * blockDim.x + threadIdx.x;
  if (e >= kE) return;
  __builtin_prefetch(bond + ((e + 4096 < kE) ? (e + 4096) : e) * 64, 0, 0);
  const int s = src[e], d = dst[e];
  const float a0 = sx[e * 2]     / denom[(size_t)d * 2];
  const float a1 = sx[e * 2 + 1] / denom[(size_t)d * 2 + 1];
  const float4* b4 = reinterpret_cast<const float4*>(bond + e * 64);
  const float4* t4 = reinterpret_cast<const float4*>(atom + (size_t)s * 64);
  float* ag0 = agg + (size_t)d * 128;
  float* ag1 = ag0 + 64;
  #pragma unroll 4
  for (int q = 0; q < 16; ++q) {
    const float4 bv = b4[q], av = t4[q];
    const float vx = av.x + bv.x, vy = av.y + bv.y, vz = av.z + bv.z, vw = av.w + bv.w;
    const int k = q * 4;
    atomicAdd(ag0 + k,     a0 * vx); atomicAdd(ag0 + k + 1, a0 * vy);
    atomicAdd(ag0 + k + 2, a0 * vz); atomicAdd(ag0 + k + 3, a0 * vw);
    atomicAdd(ag1 + k,     a1 * vx); atomicAdd(ag1 + k + 1, a1 * vy);
    atomicAdd(ag1 + k + 2, a1 * vz); atomicAdd(ag1 + k + 3, a1 * vw);
  }
}

// ---- K5: hpre[16-tile] = 0.5 * agg[16,128] @ Wc[128,64] via fp32 WMMA -----
// LDS holds Wc in pair-interleaved "fragment-ready" layout:
//   sW[p*128 + j*2 + r] = Wc[2p + r][j],  Wc[h*64+k][j] = W[k][h*64+j]
// so each B fragment (K = kb, kb+1 for column j) is ONE aligned ds_load_b64.
__global__ void __launch_bounds__(256) k_ft_wmma(const float* __restrict__ agg,
                                                 const float* __restrict__ W,
                                                 float* __restrict__ hpre,
                                                 float* __restrict__ bnsum,
                                                 float* __restrict__ bnsq) {
  __shared__ float sW[128 * 64];                     // 32 KB of the 320 KB WGP LDS
  const int tid = threadIdx.x;
  for (int i = tid; i < 128 * 64; i += 256) {
    const int kg = i >> 6, j = i & 63;               // kg = combined K, j = out col
    const int h = kg >> 6, kin = kg & 63;
    sW[(kg >> 1) * 128 + j * 2 + (kg & 1)] = W[kin * 128 + h * 64 + j];
  }
  __syncthreads();

  const int wave = tid >> 5, lane = tid & 31;        // wave32
  const int tile = blockIdx.x * 8 + wave;            // uniform per wave -> EXEC all-ones
  if (tile >= kN / 16) return;
  const int r0 = tile * 16;
  const int lm = lane & 15, lh = lane >> 4;

  v8f acc0 = {}, acc1 = {}, acc2 = {}, acc3 = {};
  // A 16x4 f32 fragment: lane<16 holds (K0,K1) of row lm; lane>=16 holds (K2,K3)
  const float* arow = agg + (size_t)(r0 + lm) * 128 + 2 * lh;
  // B fragment base for this lane: pair index p = 2*kk + lh, column lm (+16t)
  const float* bbase = sW + lh * 128 + lm * 2;
  #pragma unroll 4
  for (int kk = 0; kk < 32; ++kk) {
    const v2f a = *reinterpret_cast<const v2f*>(arow + 4 * kk);
    const float* bl = bbase + kk * 256;              // p = 2*kk + lh  ->  p*128
    const v2f b0 = *reinterpret_cast<const v2f*>(bl);
    const v2f b1 = *reinterpret_cast<const v2f*>(bl + 32);
    const v2f b2 = *reinterpret_cast<const v2f*>(bl + 64);
    const v2f b3 = *reinterpret_cast<const v2f*>(bl + 96);
    acc0 = __builtin_amdgcn_wmma_f32_16x16x4_f32(false, a, false, b0, (short)0, acc0, false, false);
    acc1 = __builtin_amdgcn_wmma_f32_16x16x4_f32(false, a, false, b1, (short)0, acc1, false, false);
    acc2 = __builtin_amdgcn_wmma_f32_16x16x4_f32(false, a, false, b2, (short)0, acc2, false, false);
    acc3 = __builtin_amdgcn_wmma_f32_16x16x4_f32(false, a, false, b3, (short)0, acc3, false, false);
  }

  auto emit = [&](v8f acc, int t) {
    const int col = t * 16 + lm;
    float csum = 0.f, csq = 0.f;
    #pragma unroll
    for (int j = 0; j < 8; ++j) {                    // C/D: VGPR j -> rows M=j (lo) / j+8 (hi)
      const float val = acc[j] * 0.5f;               // mean over 2 heads
      const int row = r0 + j + 8 * lh;
      hpre[(size_t)row * 64 + col] = val;
      csum += val; csq += val * val;
    }
    atomicAdd(bnsum + col, csum);
    atomicAdd(bnsq + col, csq);
  };
  emit(acc0, 0); emit(acc1, 1); emit(acc2, 2); emit(acc3, 3);
}

// ---- K6: BatchNorm (batch stats) + ReLU -----------------------------------
__global__ void k_bn_out(const float* __restrict__ hpre,
                         const float* __restrict__ bnsum, const float* __restrict__ bnsq,
                         const float* __restrict__ gamma, const float* __restrict__ beta,
                         float* __restrict__ out) {
  const size_t i = (size_t)blockIdx.x * blockDim.x + threadIdx.x;
  if (i >= (size_t)kN * 64) return;
  const int j = (int)(i & 63);
  const float invN = 1.0f / (float)kN;
  const float mu  = bnsum[j] * invN;
  const float var = bnsq[j] * invN - mu * mu;
  const float inv = rsqrtf(var + kEps);
  const float v = (hpre[i] - mu) * inv * gamma[j] + beta[j];
  out[i] = (v > 0.f) ? v : 0.f;
}

extern "C" void kernel_launch(void* const* d_in, const int* in_sizes, int n_in,
                              void* d_out, int out_size, void* d_ws, size_t ws_size,
                              hipStream_t stream) {
  const float* atom  = (const float*)d_in[0];
  const float* bond  = (const float*)d_in[1];
  const float* W     = (const float*)d_in[2];
  const float* attnl = (const float*)d_in[3];
  const float* attnr = (const float*)d_in[4];
  const float* gamma = (const float*)d_in[5];
  const float* beta  = (const float*)d_in[6];
  const int*   src   = (const int*)d_in[7];
  const int*   dst   = (const int*)d_in[8];
  float* out = (float*)d_out;

  // Workspace carve-up (floats): ~93 MB total.
  float* ws    = (float*)d_ws;
  float* wl    = ws;                         // 128
  float* wr    = wl + 128;                   // 128
  float* al    = wr + 128;                   // N*2
  float* er    = al + (size_t)kN * 2;        // N*2
  float* smax  = er + (size_t)kN * 2;        // N*2
  float* denom = smax + (size_t)kN * 2;      // N*2
  float* bnsum = denom + (size_t)kN * 2;     // 64
  float* bnsq  = bnsum + 64;                 // 64
  float* sx    = bnsq + 64;                  // E*2  (s, then exp(s-max) in place)
  float* agg   = sx + (size_t)kE * 2;        // N*128
  float* hpre  = agg + (size_t)kN * 128;     // N*64

  const int edgeBlocks = (kE + 255) / 256;           // 6250
  const int initBlocks = (int)(((size_t)kN * 128 + 255) / 256);
  const int nodeBlocks = (kN + 255) / 256;
  const int wmmaBlocks = (kN / 16 + 7) / 8;          // 8 waves/block, 1 tile/wave
  const int outBlocks  = (int)(((size_t)kN * 64 + 255) / 256);

  k_proj_vec  <<<1, 128, 0, stream>>>(W, attnl, attnr, wl, wr);
  k_init      <<<initBlocks, 256, 0, stream>>>(smax, denom, agg, bnsum);
  k_node      <<<nodeBlocks, 256, 0, stream>>>(atom, wl, wr, al, er);
  k_edge_score<<<edgeBlocks, 256, 0, stream>>>(bond, src, dst, wl, al, er, sx, smax);
  k_edge_exp  <<<edgeBlocks, 256, 0, stream>>>(dst, sx, smax, denom);
  k_edge_agg  <<<edgeBlocks, 256, 0, stream>>>(atom, bond, src, dst, sx, denom, agg);
  k_ft_wmma   <<<wmmaBlocks, 256, 0, stream>>>(agg, W, hpre, bnsum, bnsq);
  k_bn_out    <<<outBlocks, 256, 0, stream>>>(hpre, bnsum, bnsq, gamma, beta, out);
}